// GraphSAGEModel_62775241999123
// MI455X (gfx1250) — compile-verified
//
#include <hip/hip_runtime.h>

// ---------------------------------------------------------------------------
// GraphSAGE fused kernel for gfx1250 (MI455X), f32 WMMA 16x16x4.
// out = relu( relu( [x | mean(x[nbr])] @ W1^T + b1 ) @ W2^T + b2 )
// ---------------------------------------------------------------------------

typedef __attribute__((ext_vector_type(2))) float v2f;
typedef __attribute__((ext_vector_type(8))) float v8f;

#define N_NODES 50000
#define DEG     32
#define IN_DIM  256
#define HID     256
#define OUT_DIM 128
#define K1      (2 * IN_DIM)   // 512
#define TILE_M  16             // nodes per workgroup
#define CSTRIDE (K1 + 4)       // 516 floats: stride%64==4 -> conflict-free b64 frag loads
#define HSTRIDE (HID + 4)      // 260 floats

__global__ __launch_bounds__(256)
void sage_fused_kernel(const float* __restrict__ nf,      // [N, IN_DIM]
                       const int*   __restrict__ nidx,    // [N, DEG]
                       const float* __restrict__ W1,      // [HID, 2*IN_DIM]
                       const float* __restrict__ b1,      // [HID]
                       const float* __restrict__ W2,      // [OUT, HID]
                       const float* __restrict__ b2,      // [OUT]
                       float*       __restrict__ out)     // [N, OUT]
{
    __shared__ float comb[TILE_M][CSTRIDE];   // [x | mean] per node, f32
    __shared__ float hbuf[TILE_M][HSTRIDE];   // hidden activations, f32

    const int tid  = threadIdx.x;
    const int lane = tid & 31;
    const int wv   = tid >> 5;          // wave id 0..7
    const int base = blockIdx.x * TILE_M;

    // ---------------- Phase A1: copy self features into comb[:, 0:256] -----
    {
        int row  = tid >> 4;            // 0..15
        int sub  = tid & 15;
        int node = base + row;
        if (node >= N_NODES) node = N_NODES - 1;
        const float4* src = (const float4*)nf + node * (IN_DIM / 4);
        float4* dst = (float4*)&comb[row][0];
#pragma unroll
        for (int j = 0; j < 4; ++j)
            dst[sub + 16 * j] = src[sub + 16 * j];
    }

    // ---------------- Phase A2: mean over 32 neighbors into comb[:, 256:512]
    // Each wave handles 2 rows; lanes stripe the 256-wide feature dim as
    // 2 float4 chunks; neighbor ids broadcast from a per-lane register.
#pragma unroll
    for (int rr = 0; rr < 2; ++rr) {
        const int row  = wv * 2 + rr;
        int node = base + row;
        if (node >= N_NODES) node = N_NODES - 1;
        const int myidx = nidx[node * DEG + lane];  // lane d holds neighbor d

        float a0x = 0.f, a0y = 0.f, a0z = 0.f, a0w = 0.f;
        float a1x = 0.f, a1y = 0.f, a1z = 0.f, a1w = 0.f;
        const float4* nf4 = (const float4*)nf;
#pragma unroll
        for (int d = 0; d < DEG; ++d) {
            const int nbr = __shfl(myidx, d, 32);
            const float4* srow = nf4 + nbr * (IN_DIM / 4);
            if (d + 1 < DEG) {   // pull next random row L2 -> L0 early
                const int nn = __shfl(myidx, d + 1, 32);
                __builtin_prefetch((const void*)(nf4 + nn * (IN_DIM / 4) + lane), 0, 1);
            }
            const float4 x0 = srow[lane];
            const float4 x1 = srow[lane + 32];
            a0x += x0.x; a0y += x0.y; a0z += x0.z; a0w += x0.w;
            a1x += x1.x; a1y += x1.y; a1z += x1.z; a1w += x1.w;
        }
        const float s = 1.0f / (float)DEG;
        float4* drow = (float4*)&comb[row][IN_DIM];
        float4 m0; m0.x = a0x * s; m0.y = a0y * s; m0.z = a0z * s; m0.w = a0w * s;
        float4 m1; m1.x = a1x * s; m1.y = a1y * s; m1.z = a1z * s; m1.w = a1w * s;
        drow[lane]      = m0;
        drow[lane + 32] = m1;
    }
    __syncthreads();

    // WMMA fragment lane mapping (f32 16x16x4):
    //   A: lane L holds row (L&15), k-offsets 2*(L>>4)+{0,1}
    //   B: lane L holds col (L&15), k-offsets 2*(L>>4)+{0,1}
    //   C/D: lane L holds col (L&15), rows r + 8*(L>>4) in VGPR r
    const int mrow = lane & 15;
    const int koff = (lane >> 4) << 1;   // 0 or 2
    const int rsh  = (lane >> 4) << 3;   // 0 or 8

    // ---------------- Phase B: h = relu(comb @ W1^T + b1), K=512 -----------
    {
        const int n0 = wv * 16 + mrow;          // n-tile wv
        const int n1 = (wv + 8) * 16 + mrow;    // n-tile wv+8
        const float* w1r0 = W1 + n0 * K1 + koff;
        const float* w1r1 = W1 + n1 * K1 + koff;
        const float* arow = &comb[mrow][koff];

        v8f c0 = {}; v8f c1 = {};
#pragma unroll 4
        for (int k = 0; k < K1; k += 4) {
            v2f a  = *(const v2f*)(arow + k);
            v2f bb0 = *(const v2f*)(w1r0 + k);
            v2f bb1 = *(const v2f*)(w1r1 + k);
            c0 = __builtin_amdgcn_wmma_f32_16x16x4_f32(false, a, false, bb0,
                                                       (short)0, c0, false, false);
            c1 = __builtin_amdgcn_wmma_f32_16x16x4_f32(false, a, false, bb1,
                                                       (short)0, c1, false, false);
        }
        const float bias0 = b1[wv * 16 + mrow];
        const float bias1 = b1[(wv + 8) * 16 + mrow];
#pragma unroll
        for (int r = 0; r < 8; ++r) {
            float h0 = c0[r] + bias0; h0 = h0 > 0.f ? h0 : 0.f;
            float h1 = c1[r] + bias1; h1 = h1 > 0.f ? h1 : 0.f;
            hbuf[r + rsh][wv * 16 + mrow]       = h0;
            hbuf[r + rsh][(wv + 8) * 16 + mrow] = h1;
        }
    }
    __syncthreads();

    // ---------------- Phase C: out = relu(h @ W2^T + b2), K=256 ------------
    {
        const int n = wv * 16 + mrow;           // output column 0..127
        const float* w2r  = W2 + n * HID + koff;
        const float* arow = &hbuf[mrow][koff];

        v8f c = {};
#pragma unroll 4
        for (int k = 0; k < HID; k += 4) {
            v2f a  = *(const v2f*)(arow + k);
            v2f bb = *(const v2f*)(w2r + k);
            c = __builtin_amdgcn_wmma_f32_16x16x4_f32(false, a, false, bb,
                                                      (short)0, c, false, false);
        }
        const float bias = b2[n];
#pragma unroll
        for (int r = 0; r < 8; ++r) {
            const int node = base + r + rsh;
            if (node < N_NODES) {
                float v = c[r] + bias; v = v > 0.f ? v : 0.f;
                out[node * OUT_DIM + n] = v;
            }
        }
    }
}

extern "C" void kernel_launch(void* const* d_in, const int* in_sizes, int n_in,
                              void* d_out, int out_size, void* d_ws, size_t ws_size,
                              hipStream_t stream) {
    const float* nf   = (const float*)d_in[0];
    const int*   nidx = (const int*)  d_in[1];
    const float* W1   = (const float*)d_in[2];
    const float* b1   = (const float*)d_in[3];
    const float* W2   = (const float*)d_in[4];
    const float* b2   = (const float*)d_in[5];
    float* out = (float*)d_out;

    const int blocks = (N_NODES + TILE_M - 1) / TILE_M;   // 3125
    hipLaunchKernelGGL(sage_fused_kernel, dim3(blocks), dim3(256), 0, stream,
                       nf, nidx, W1, b1, W2, b2, out);
}